// PointRefiner_75350906241274
// MI455X (gfx1250) — compile-verified
//
#include <hip/hip_runtime.h>
#include <hip/hip_bf16.h>

using u32 = unsigned int;
using u64 = unsigned long long;

#define BATCHES   16
#define IMG_W     1024
#define NELEM     (1024 * 1024)     // elements per batch
#define CHUNK     8192              // elements per block per sweep
#define BLOCKS_X  (NELEM / CHUNK)   // 128
#define NPROB     48                // 3 scores x 16 batches, pid = s*16 + b
#define BLOCK     256
#define PASSES    8

// output layout (floats): Omega_K 16x512x2 | Omega_L 16x128x2 | Omega_I 16x128x2
#define OUT_L_OFF 16384
#define OUT_I_OFF 20480

// ---- optional gfx1250 async global->LDS path -------------------------------
#if defined(__has_builtin)
#if __has_builtin(__builtin_amdgcn_global_load_async_to_lds_b64) && \
    __has_builtin(__builtin_amdgcn_s_wait_asynccnt)
#define USE_ASYNC_LDS 1
#endif
#endif

typedef int v2i __attribute__((ext_vector_type(2)));
typedef __attribute__((address_space(1))) v2i* as1_v2i;   // global b64 payload
typedef __attribute__((address_space(3))) v2i* as3_v2i;   // LDS b64 payload

// ---- key construction ------------------------------------------------------
// key = (float_bits(score) << 32) | ~index  : unique, descending key order ==
// descending score then ascending index (matches jax.lax.top_k tie-breaking).
__device__ __forceinline__ void make_keys(float pv, u32 idx,
                                          u64& ku, u64& kp, u64& kq) {
  u64 inv = (u64)(u32)(~idx);
  float un = 1.0f - fabsf(2.0f * pv - 1.0f);   // uncertainty
  float qv = 1.0f - pv;                        // background confidence
  ku = ((u64)__float_as_uint(un) << 32) | inv;
  kp = ((u64)__float_as_uint(pv) << 32) | inv;
  kq = ((u64)__float_as_uint(qv) << 32) | inv;
}

// Wave32-aggregated LDS histogram add: one ds_add per distinct bin per wave.
__device__ __forceinline__ void hist_add_wave(u32* lh, u32 bin, bool match) {
  u64 pend = __ballot(match ? 1 : 0);
  u32 lane = threadIdx.x & 31u;
  while (pend) {
    u32 leader = (u32)__builtin_ctzll(pend);
    u32 lbin = (u32)__shfl((int)bin, (int)leader, 32);
    u64 same = __ballot((match && (bin == lbin)) ? 1 : 0);
    if (lane == leader) atomicAdd(&lh[lbin], (u32)__popcll(same));
    pend &= ~same;
    match = match && (bin != lbin);
  }
}

// ---- kernel 0: init workspace ----------------------------------------------
__global__ void k_init(u32* __restrict__ need, u64* __restrict__ prefix,
                       u32* __restrict__ cnt, u32* __restrict__ hist) {
  int pid = blockIdx.x, tid = threadIdx.x;
  hist[pid * 256 + tid] = 0u;
  if (tid == 0) {
    prefix[pid] = 0ull;
    cnt[pid] = 0u;
    need[pid] = (pid < 16) ? 512u : 128u;
  }
}

// ---- kernel 1: one radix histogram pass over all data ----------------------
__global__ void k_hist(const float* __restrict__ p, const u64* __restrict__ prefix,
                       u32* __restrict__ hist, int shift, u64 hm) {
  __shared__ u32 lh[768];
  const int tid = threadIdx.x;
  const int b = blockIdx.y;
  for (int i = tid; i < 768; i += BLOCK) lh[i] = 0u;
  const u64 pref0 = prefix[b];
  const u64 pref1 = prefix[16 + b];
  const u64 pref2 = prefix[32 + b];
  __syncthreads();

  const float4* p4 = (const float4*)(p + (size_t)b * NELEM + (size_t)blockIdx.x * CHUNK);
  const u32 base_idx = (u32)blockIdx.x * CHUNK;

#pragma unroll
  for (int it = 0; it < 8; ++it) {
    int vec = it * BLOCK + tid;
    float4 v = p4[vec];
    if (it + 1 < 8) __builtin_prefetch((const void*)(p4 + vec + BLOCK), 0, 0);
    float vals[4] = {v.x, v.y, v.z, v.w};
    u32 idx0 = base_idx + (u32)vec * 4u;
#pragma unroll
    for (int c = 0; c < 4; ++c) {
      u64 ku, kp, kq;
      make_keys(vals[c], idx0 + (u32)c, ku, kp, kq);
      hist_add_wave(lh,       (u32)((ku >> shift) & 255u), (ku & hm) == pref0);
      hist_add_wave(lh + 256, (u32)((kp >> shift) & 255u), (kp & hm) == pref1);
      hist_add_wave(lh + 512, (u32)((kq >> shift) & 255u), (kq & hm) == pref2);
    }
  }
  __syncthreads();
  for (int i = tid; i < 768; i += BLOCK) {
    u32 c = lh[i];
    if (c) atomicAdd(&hist[(u32)(((i >> 8) * 16 + b) * 256 + (i & 255))], c);
  }
}

// ---- kernel 2: pick digit for each problem, reset histogram ----------------
__global__ void k_scan(u32* __restrict__ need, u64* __restrict__ prefix,
                       u32* __restrict__ hist, int shift) {
  __shared__ u32 lh[256];
  int pid = blockIdx.x, tid = threadIdx.x;
  lh[tid] = hist[pid * 256 + tid];
  __syncthreads();
  hist[pid * 256 + tid] = 0u;            // ready for next pass
  if (tid == 0) {
    u32 nd = need[pid];
    u32 cum = 0; int ch = 0;
    for (int bn = 255; bn >= 0; --bn) {
      u32 c = lh[bn];
      if (cum + c >= nd) { ch = bn; nd -= cum; break; }
      cum += c;
    }
    need[pid] = nd;
    prefix[pid] |= ((u64)(u32)ch) << shift;
  }
}

// ---- kernel 3: compact — emit the exactly-k keys >= kth key ----------------
__global__ void k_compact(const float* __restrict__ p, const u64* __restrict__ kth,
                          u32* __restrict__ cnt, u64* __restrict__ cand) {
  const int tid = threadIdx.x;
  const int b = blockIdx.y;
  const u64 k0 = kth[b], k1 = kth[16 + b], k2 = kth[32 + b];
  const float4* p4 = (const float4*)(p + (size_t)b * NELEM + (size_t)blockIdx.x * CHUNK);
  const u32 base_idx = (u32)blockIdx.x * CHUNK;

#pragma unroll
  for (int it = 0; it < 8; ++it) {
    int vec = it * BLOCK + tid;
    float4 v = p4[vec];
    float vals[4] = {v.x, v.y, v.z, v.w};
    u32 idx0 = base_idx + (u32)vec * 4u;
#pragma unroll
    for (int c = 0; c < 4; ++c) {
      u64 ku, kp, kq;
      make_keys(vals[c], idx0 + (u32)c, ku, kp, kq);
      if (ku >= k0) { u32 pos = atomicAdd(&cnt[b],      1u); cand[(size_t)b        * 512 + pos] = ku; }
      if (kp >= k1) { u32 pos = atomicAdd(&cnt[16 + b], 1u); cand[(size_t)(16 + b) * 512 + pos] = kp; }
      if (kq >= k2) { u32 pos = atomicAdd(&cnt[32 + b], 1u); cand[(size_t)(32 + b) * 512 + pos] = kq; }
    }
  }
}

// ---- kernel 4: per-problem bitonic sort (desc) + coordinate emit -----------
__global__ void k_sort_emit(const u64* __restrict__ cand, float* __restrict__ out) {
  __shared__ u64 sk[512];
  const int pid = blockIdx.x;
  const int tid = threadIdx.x;
  const u32 k = (pid < 16) ? 512u : 128u;
  const u64* src = cand + (size_t)pid * 512;

#ifdef USE_ASYNC_LDS
  for (u32 j = tid; j < k; j += BLOCK) {
    __builtin_amdgcn_global_load_async_to_lds_b64(
        (as1_v2i)(u64)(const void*)(src + j),
        (as3_v2i)(u32)(u64)(void*)(&sk[j]),
        0, 0);
  }
  __builtin_amdgcn_s_wait_asynccnt(0);
#else
  for (u32 j = tid; j < k; j += BLOCK) sk[j] = src[j];
#endif
  __syncthreads();

  // bitonic sort, descending (all pairs within a stage are disjoint)
  for (u32 sz = 2; sz <= k; sz <<= 1) {
    for (u32 st = sz >> 1; st > 0; st >>= 1) {
      for (u32 i = tid; i < k; i += BLOCK) {
        u32 j = i ^ st;
        if (j > i) {
          u64 a = sk[i], bb = sk[j];
          bool up = ((i & sz) == 0);     // "up" half of descending network
          if (up ? (a < bb) : (a > bb)) { sk[i] = bb; sk[j] = a; }
        }
      }
      __syncthreads();
    }
  }

  const int s = pid / 16, b = pid % 16;
  float* obase = (s == 0) ? (out + (size_t)b * 1024)
               : (s == 1) ? (out + OUT_L_OFF + (size_t)b * 256)
                          : (out + OUT_I_OFF + (size_t)b * 256);
  for (u32 j = tid; j < k; j += BLOCK) {
    u32 idx = ~(u32)(sk[j] & 0xFFFFFFFFull);
    obase[2 * j]     = (float)(idx & (IMG_W - 1));  // x
    obase[2 * j + 1] = (float)(idx >> 10);          // y
  }
}

// ---- host launch -----------------------------------------------------------
extern "C" void kernel_launch(void* const* d_in, const int* in_sizes, int n_in,
                              void* d_out, int out_size, void* d_ws, size_t ws_size,
                              hipStream_t stream) {
  const float* p = (const float*)d_in[0];   // (16,1,1024,1024) fp32
  // d_in[1] (feature_map) is unused by the reference.
  float* out = (float*)d_out;               // 24576 floats

  char* ws = (char*)d_ws;                   // ~247 KB used
  u64* st_prefix = (u64*)(ws);              // 48 * 8   = 384 B
  u32* st_need   = (u32*)(ws + 384);        // 48 * 4   = 192 B
  u32* cnt       = (u32*)(ws + 576);        // 48 * 4   = 192 B
  u32* hist      = (u32*)(ws + 768);        // 48*256*4 = 49152 B
  u64* cand      = (u64*)(ws + 49920);      // 48*512*8 = 196608 B

  k_init<<<NPROB, BLOCK, 0, stream>>>(st_need, st_prefix, cnt, hist);

  for (int t = 0; t < PASSES; ++t) {
    int shift = 56 - 8 * t;
    u64 hm = (t == 0) ? 0ull : (~0ull << (shift + 8));
    k_hist<<<dim3(BLOCKS_X, BATCHES), BLOCK, 0, stream>>>(p, st_prefix, hist, shift, hm);
    k_scan<<<NPROB, BLOCK, 0, stream>>>(st_need, st_prefix, hist, shift);
  }

  k_compact<<<dim3(BLOCKS_X, BATCHES), BLOCK, 0, stream>>>(p, st_prefix, cnt, cand);
  k_sort_emit<<<NPROB, BLOCK, 0, stream>>>(cand, out);
}